// PushoverGNN_86921548136602
// MI455X (gfx1250) — compile-verified
//
#include <hip/hip_runtime.h>
#include <math.h>

// ---------------- problem constants (match reference) ----------------
#define NN   20000
#define EE   320000
#define BB   64
#define NDD  6
#define NDP  8        // padded K for layer 0
#define EDD  16
#define GDD  8
#define HDD  256      // H*O
#define OUTD 100
#define ZDD  (HDD + GDD)   // 264
#define NEG_SLOPE 0.2f
#define BN_EPS 1e-5f

typedef float v2f __attribute__((ext_vector_type(2)));
typedef float v8f __attribute__((ext_vector_type(8)));

// ---------------- helpers ----------------
__device__ __forceinline__ float atomicMaxFloat(float* addr, float value) {
  if (value >= 0.0f)
    return __int_as_float(atomicMax((int*)addr, __float_as_int(value)));
  else
    return __uint_as_float(atomicMin((unsigned int*)addr, __float_as_uint(value)));
}

__global__ void fill_f32(float* __restrict__ p, float v, int n) {
  int i = blockIdx.x * blockDim.x + threadIdx.x;
  if (i < n) p[i] = v;
}

// pad x [N,6] -> [N,8] with zeros
__global__ void pad_x_k(const float* __restrict__ x, float* __restrict__ xpad, int Nn) {
  int i = blockIdx.x * blockDim.x + threadIdx.x;
  if (i >= Nn * NDP) return;
  const int n = i >> 3, j = i & 7;
  xpad[i] = (j < NDD) ? x[(size_t)n * NDD + j] : 0.0f;
}

// BT[n*Kpad + k] = (k<K) ? B[k*N + n] : 0   (weights are tiny; cost negligible)
__global__ void transpose_pad_k(const float* __restrict__ B, float* __restrict__ BT,
                                int K, int Kpad, int N) {
  int i = blockIdx.x * blockDim.x + threadIdx.x;
  if (i >= N * Kpad) return;
  const int n = i / Kpad, k = i % Kpad;
  BT[i] = (k < K) ? B[(size_t)k * N + n] : 0.0f;
}

// ---------------- FAST fp32 WMMA GEMM ----------------
// C[M,N] = A[M,K] @ B[K,N], with B supplied TRANSPOSED as BT[N,K].
// Requirements: M%16==0, N%64==0, K%4==0, K even (for b64-aligned loads).
// One wave computes a 16x64 strip: A fragment loaded once per K-step (b64),
// reused across 4 v_wmma_f32_16x16x4_f32; B fragments are contiguous b64 loads.
// Fragment layouts per ISA (32-bit A 16x4 / B 4x16 / C 16x16 tables):
//   lane = 16h + r;  a = {A[m0+r, ka], A[m0+r, ka+1]},  ka = k0 + 2h
//   b_t  = {B[ka, n0+16t+r], B[ka+1, n0+16t+r]} = BT[(n0+16t+r)*K + ka .. +1]
//   acc_t[v] -> C[m0 + v + 8h, n0 + 16t + r]
__global__ void gemm_wmma_fast(const float* __restrict__ A, const float* __restrict__ BT,
                               const float* __restrict__ bias, float* __restrict__ C,
                               int M, int K, int N, int relu) {
  const int wave = threadIdx.x >> 5;
  const int lane = threadIdx.x & 31;
  const int tilesN = N >> 6;                 // 64-wide strips
  const int tile   = blockIdx.x * (blockDim.x >> 5) + wave;   // wave-uniform
  if (tile >= (M >> 4) * tilesN) return;     // whole wave exits together
  const int m0 = (tile / tilesN) << 4;
  const int n0 = (tile % tilesN) << 6;
  const int h  = lane >> 4;
  const int r  = lane & 15;

  const float* __restrict__ Arow = A + (size_t)(m0 + r) * K + 2 * h;
  const float* __restrict__ Bc0  = BT + (size_t)(n0 + r)      * K + 2 * h;
  const float* __restrict__ Bc1  = BT + (size_t)(n0 + 16 + r) * K + 2 * h;
  const float* __restrict__ Bc2  = BT + (size_t)(n0 + 32 + r) * K + 2 * h;
  const float* __restrict__ Bc3  = BT + (size_t)(n0 + 48 + r) * K + 2 * h;

  v8f acc0 = {}, acc1 = {}, acc2 = {}, acc3 = {};
  #pragma unroll 2
  for (int k0 = 0; k0 < K; k0 += 4) {
    const v2f a  = *(const v2f*)(Arow + k0);
    const v2f b0 = *(const v2f*)(Bc0 + k0);
    const v2f b1 = *(const v2f*)(Bc1 + k0);
    const v2f b2 = *(const v2f*)(Bc2 + k0);
    const v2f b3 = *(const v2f*)(Bc3 + k0);
    acc0 = __builtin_amdgcn_wmma_f32_16x16x4_f32(false, a, false, b0, 0, acc0, false, false);
    acc1 = __builtin_amdgcn_wmma_f32_16x16x4_f32(false, a, false, b1, 0, acc1, false, false);
    acc2 = __builtin_amdgcn_wmma_f32_16x16x4_f32(false, a, false, b2, 0, acc2, false, false);
    acc3 = __builtin_amdgcn_wmma_f32_16x16x4_f32(false, a, false, b3, 0, acc3, false, false);
  }

  float* __restrict__ Crow = C + (size_t)(m0 + 8 * h) * N + n0 + r;
  #pragma unroll
  for (int v = 0; v < 8; ++v) {
    float v0 = acc0[v], v1 = acc1[v], v2 = acc2[v], v3 = acc3[v];
    if (bias) {
      v0 += bias[n0 + r]; v1 += bias[n0 + 16 + r];
      v2 += bias[n0 + 32 + r]; v3 += bias[n0 + 48 + r];
    }
    if (relu) {
      v0 = v0 > 0.f ? v0 : 0.f; v1 = v1 > 0.f ? v1 : 0.f;
      v2 = v2 > 0.f ? v2 : 0.f; v3 = v3 > 0.f ? v3 : 0.f;
    }
    float* cp = Crow + (size_t)v * N;
    cp[0] = v0; cp[16] = v1; cp[32] = v2; cp[48] = v3;
  }
}

// ---------------- generic (bounds-checked) WMMA GEMM: only for the tiny 64x100 head ----------------
__global__ void gemm_wmma_gen(const float* __restrict__ A, const float* __restrict__ Bm,
                              const float* __restrict__ bias, float* __restrict__ C,
                              int M, int K, int N, int relu) {
  const int wave = threadIdx.x >> 5;
  const int lane = threadIdx.x & 31;
  const int tilesN = (N + 15) >> 4;
  const int tilesM = (M + 15) >> 4;
  const int tile = blockIdx.x * (blockDim.x >> 5) + wave;
  if (tile >= tilesM * tilesN) return;
  const int m0 = (tile / tilesN) << 4;
  const int n0 = (tile % tilesN) << 4;
  const int h  = lane >> 4;
  const int r  = lane & 15;
  const int arow = m0 + r;
  const int bcol = n0 + r;
  const bool rowOK = arow < M;
  const bool colOK = bcol < N;

  v8f acc = {};
  for (int k0 = 0; k0 < K; k0 += 4) {
    const int ka = k0 + 2 * h;
    v2f a, b;
    a.x = (rowOK && ka     < K) ? A[(size_t)arow * K + ka]     : 0.0f;
    a.y = (rowOK && ka + 1 < K) ? A[(size_t)arow * K + ka + 1] : 0.0f;
    b.x = (colOK && ka     < K) ? Bm[(size_t)ka * N + bcol]       : 0.0f;
    b.y = (colOK && ka + 1 < K) ? Bm[(size_t)(ka + 1) * N + bcol] : 0.0f;
    acc = __builtin_amdgcn_wmma_f32_16x16x4_f32(false, a, false, b, 0, acc, false, false);
  }
  #pragma unroll
  for (int v = 0; v < 8; ++v) {
    const int row = m0 + v + 8 * h;
    const int col = n0 + r;
    if (row < M && col < N) {
      float val = acc[v];
      if (bias) val += bias[col];
      if (relu) val = val > 0.0f ? val : 0.0f;
      C[(size_t)row * N + col] = val;
    }
  }
}

// ---------------- edge score: fused ee-GEMM + gather + leakyrelu + att dot + seg-max ----------------
__global__ void edge_score_k(const int* __restrict__ src, const int* __restrict__ dst,
                             const float* __restrict__ ea,   // [E,16]
                             const float* __restrict__ We,   // [16,256]
                             const float* __restrict__ att,  // [2*128] flat
                             const float* __restrict__ xl, const float* __restrict__ xr,
                             float* __restrict__ score,      // [E,2]
                             float* __restrict__ mbuf,       // [N,2], pre-filled -inf
                             int E_) {
  __shared__ float sWe[EDD * HDD];   // 16 KB
  __shared__ float sAtt[HDD];        // 1 KB
  for (int i = threadIdx.x; i < EDD * HDD; i += blockDim.x) sWe[i] = We[i];
  for (int i = threadIdx.x; i < HDD; i += blockDim.x) sAtt[i] = att[i];
  __syncthreads();

  const int lane = threadIdx.x & 31;
  const int gw   = blockIdx.x * (blockDim.x >> 5) + (threadIdx.x >> 5);
  const int nw   = gridDim.x * (blockDim.x >> 5);
  const int c0   = lane * 8;

  for (int e = gw; e < E_; e += nw) {
    const int s = src[e], d = dst[e];
    float ev[EDD];
    #pragma unroll
    for (int k = 0; k < EDD; ++k) ev[k] = ea[(size_t)e * EDD + k];  // wave-broadcast loads
    float acc = 0.0f;
    #pragma unroll
    for (int j = 0; j < 8; ++j) {
      const int c = c0 + j;
      float eec = 0.0f;
      #pragma unroll
      for (int k = 0; k < EDD; ++k) eec += ev[k] * sWe[k * HDD + c];
      float v = xl[(size_t)s * HDD + c] + xr[(size_t)d * HDD + c] + eec;
      v = v > 0.0f ? v : NEG_SLOPE * v;
      acc += sAtt[c] * v;
    }
    #pragma unroll
    for (int off = 1; off < 16; off <<= 1) acc += __shfl_xor(acc, off, 32);
    if ((lane & 15) == 0) {
      const int hh = lane >> 4;
      score[(size_t)e * 2 + hh] = acc;
      atomicMaxFloat(&mbuf[(size_t)d * 2 + hh], acc);
    }
  }
}

// ---------------- p = exp(score - m[dst]); denom[dst] += p ----------------
__global__ void edge_exp_k(const int* __restrict__ dst, const float* __restrict__ score,
                           const float* __restrict__ mbuf, float* __restrict__ p,
                           float* __restrict__ denom, int E_) {
  int i = blockIdx.x * blockDim.x + threadIdx.x;
  if (i >= E_ * 2) return;
  const int e = i >> 1, hh = i & 1;
  const int d = dst[e];
  const float pv = __expf(score[i] - mbuf[(size_t)d * 2 + hh]);
  p[i] = pv;
  atomicAdd(&denom[(size_t)d * 2 + hh], pv);
}

// ---------------- out[dst] += (p/denom[dst]) * xl[src] ----------------
__global__ void edge_scatter_k(const int* __restrict__ src, const int* __restrict__ dst,
                               const float* __restrict__ p, const float* __restrict__ denom,
                               const float* __restrict__ xl, float* __restrict__ hout, int E_) {
  const int lane = threadIdx.x & 31;
  const int gw   = blockIdx.x * (blockDim.x >> 5) + (threadIdx.x >> 5);
  const int nw   = gridDim.x * (blockDim.x >> 5);
  const int c0   = lane * 8;
  const int hh   = lane >> 4;
  for (int e = gw; e < E_; e += nw) {
    const int s = src[e], d = dst[e];
    const float alpha = p[(size_t)e * 2 + hh] / denom[(size_t)d * 2 + hh];
    #pragma unroll
    for (int j = 0; j < 8; ++j) {
      const int c = c0 + j;
      atomicAdd(&hout[(size_t)d * HDD + c], alpha * xl[(size_t)s * HDD + c]);
    }
  }
}

// ---------------- BN stats: per-channel mean/var over node axis ----------------
__global__ void bn_stats_k(const float* __restrict__ hout, const float* __restrict__ bc,
                           float* __restrict__ mu, float* __restrict__ var, int Nn) {
  const int c = blockIdx.x;
  const float add = bc[c];
  float s = 0.0f, s2 = 0.0f;
  for (int i = threadIdx.x; i < Nn; i += blockDim.x) {
    const float v = hout[(size_t)i * HDD + c] + add;
    s += v; s2 += v * v;
  }
  __shared__ float ss[256], ss2[256];
  ss[threadIdx.x] = s; ss2[threadIdx.x] = s2;
  __syncthreads();
  for (int o = blockDim.x >> 1; o > 0; o >>= 1) {
    if (threadIdx.x < (unsigned)o) { ss[threadIdx.x] += ss[threadIdx.x + o]; ss2[threadIdx.x] += ss2[threadIdx.x + o]; }
    __syncthreads();
  }
  if (threadIdx.x == 0) {
    const float mean = ss[0] / (float)Nn;
    mu[c]  = mean;
    var[c] = ss2[0] / (float)Nn - mean * mean;
  }
}

__global__ void bn_relu_k(const float* __restrict__ hout, const float* __restrict__ bc,
                          const float* __restrict__ mu, const float* __restrict__ var,
                          const float* __restrict__ gamma, const float* __restrict__ beta,
                          float* __restrict__ hbuf, int total) {
  int i = blockIdx.x * blockDim.x + threadIdx.x;
  if (i >= total) return;
  const int c = i & (HDD - 1);
  const float v = hout[i] + bc[c];
  float y = gamma[c] * (v - mu[c]) * rsqrtf(var[c] + BN_EPS) + beta[c];
  hbuf[i] = y > 0.0f ? y : 0.0f;
}

// ---------------- mean pool by batch + build MLP input ----------------
__global__ void pool_accum_k(const float* __restrict__ hbuf, const int* __restrict__ batch,
                             float* __restrict__ gsum, float* __restrict__ cnt, int Nn) {
  int i = blockIdx.x * blockDim.x + threadIdx.x;
  if (i >= Nn * HDD) return;
  const int n = i >> 8, c = i & (HDD - 1);
  const int b = batch[n];
  atomicAdd(&gsum[(size_t)b * HDD + c], hbuf[i]);
  if (c == 0) atomicAdd(&cnt[b], 1.0f);
}

__global__ void build_z_k(const float* __restrict__ gsum, const float* __restrict__ cnt,
                          const float* __restrict__ gfeat, float* __restrict__ z, int Bn) {
  int i = blockIdx.x * blockDim.x + threadIdx.x;
  if (i >= Bn * ZDD) return;
  const int b = i / ZDD, c = i % ZDD;
  z[i] = (c < HDD) ? gsum[(size_t)b * HDD + c] / cnt[b]
                   : gfeat[(size_t)b * GDD + (c - HDD)];
}

// ---------------- host-side helpers ----------------
static inline void launch_fill(float* p, float v, int n, hipStream_t s) {
  fill_f32<<<(n + 255) / 256, 256, 0, s>>>(p, v, n);
}
static inline void launch_transpose(const float* B, float* BT, int K, int Kpad, int N,
                                    hipStream_t s) {
  transpose_pad_k<<<(N * Kpad + 255) / 256, 256, 0, s>>>(B, BT, K, Kpad, N);
}
// fast GEMM: M%16==0, N%64==0, K%4==0, BT[N,K] pre-transposed
static inline void launch_gemm_fast(const float* A, const float* BT, const float* bias,
                                    float* C, int M, int K, int Nc, int relu, hipStream_t s) {
  const int tiles  = (M / 16) * (Nc / 64);
  const int blocks = (tiles + 7) / 8;
  gemm_wmma_fast<<<blocks, 256, 0, s>>>(A, BT, bias, C, M, K, Nc, relu);
}

extern "C" void kernel_launch(void* const* d_in, const int* in_sizes, int n_in,
                              void* d_out, int out_size, void* d_ws, size_t ws_size,
                              hipStream_t stream) {
  (void)in_sizes; (void)n_in; (void)out_size; (void)ws_size;

  const float* x          = (const float*)d_in[0];
  const int*   edge_index = (const int*)  d_in[1];
  const float* edge_attr  = (const float*)d_in[2];
  const int*   batch      = (const int*)  d_in[3];
  const float* gfeat      = (const float*)d_in[4];
  const int*   src = edge_index;
  const int*   dst = edge_index + EE;
  const float* W1 = (const float*)d_in[26];
  const float* b1 = (const float*)d_in[27];
  const float* W2 = (const float*)d_in[28];
  const float* b2 = (const float*)d_in[29];
  const float* W3 = (const float*)d_in[30];
  const float* b3 = (const float*)d_in[31];
  float* out = (float*)d_out;

  // workspace carve-up (fp32)
  float* W = (float*)d_ws;
  size_t o = 0;
  float* xl    = W + o; o += (size_t)NN * HDD;
  float* xr    = W + o; o += (size_t)NN * HDD;
  float* hbuf  = W + o; o += (size_t)NN * HDD;
  float* hout  = W + o; o += (size_t)NN * HDD;
  float* score = W + o; o += (size_t)EE * 2;
  float* pbuf  = W + o; o += (size_t)EE * 2;
  float* mbuf  = W + o; o += (size_t)NN * 2;
  float* denom = W + o; o += (size_t)NN * 2;
  float* mu    = W + o; o += HDD;
  float* var   = W + o; o += HDD;
  float* gsum  = W + o; o += (size_t)BB * HDD;
  float* cnt   = W + o; o += BB;
  float* zbuf  = W + o; o += (size_t)BB * ZDD;
  float* z1    = W + o; o += (size_t)BB * 256;
  float* z2    = W + o; o += (size_t)BB * 128;
  float* xpad  = W + o; o += (size_t)NN * NDP;       // x padded to K=8
  float* BT1   = W + o; o += (size_t)HDD * ZDD;      // transposed weight scratch A
  float* BT2   = W + o; o += (size_t)HDD * ZDD;      // transposed weight scratch B

  // pad layer-0 input once (K=6 -> 8 so the fast kernel's b64 loads stay aligned)
  pad_x_k<<<(NN * NDP + 255) / 256, 256, 0, stream>>>(x, xpad, NN);

  const float* h    = xpad;
  int          din  = NDP;   // padded K for layer 0
  int          kraw = NDD;   // real K of the weights

  for (int l = 0; l < 3; ++l) {
    const int base = 5 + l * 7;
    const float* Wl    = (const float*)d_in[base + 0];
    const float* Wr    = (const float*)d_in[base + 1];
    const float* We    = (const float*)d_in[base + 2];
    const float* att   = (const float*)d_in[base + 3];
    const float* bc    = (const float*)d_in[base + 4];
    const float* gamma = (const float*)d_in[base + 5];
    const float* beta  = (const float*)d_in[base + 6];

    // node transforms (fast WMMA path; weights pre-transposed/zero-padded)
    launch_transpose(Wl, BT1, kraw, din, HDD, stream);
    launch_transpose(Wr, BT2, kraw, din, HDD, stream);
    launch_gemm_fast(h, BT1, nullptr, xl, NN, din, HDD, 0, stream);
    launch_gemm_fast(h, BT2, nullptr, xr, NN, din, HDD, 0, stream);

    // reset segment accumulators
    launch_fill(mbuf, -INFINITY, NN * 2, stream);
    launch_fill(denom, 0.0f, NN * 2, stream);
    launch_fill(hout, 0.0f, NN * HDD, stream);

    // attention pipeline (L2-resident gathers / atomics)
    edge_score_k<<<5000, 256, 0, stream>>>(src, dst, edge_attr, We, att, xl, xr,
                                           score, mbuf, EE);
    edge_exp_k<<<(EE * 2 + 255) / 256, 256, 0, stream>>>(dst, score, mbuf, pbuf, denom, EE);
    edge_scatter_k<<<5000, 256, 0, stream>>>(src, dst, pbuf, denom, xl, hout, EE);

    // bias + batchnorm(train stats) + relu
    bn_stats_k<<<HDD, 256, 0, stream>>>(hout, bc, mu, var, NN);
    bn_relu_k<<<(NN * HDD + 255) / 256, 256, 0, stream>>>(hout, bc, mu, var, gamma, beta,
                                                          hbuf, NN * HDD);
    h    = hbuf;
    din  = HDD;
    kraw = HDD;
  }

  // mean pool per graph, concat global features
  launch_fill(gsum, 0.0f, BB * HDD, stream);
  launch_fill(cnt, 0.0f, BB, stream);
  pool_accum_k<<<(NN * HDD + 255) / 256, 256, 0, stream>>>(hbuf, batch, gsum, cnt, NN);
  build_z_k<<<(BB * ZDD + 255) / 256, 256, 0, stream>>>(gsum, cnt, gfeat, zbuf, BB);

  // MLP head: [64,264]@[264,256] relu, [64,256]@[256,128] relu (fast path),
  // final [64,128]@[128,100] via generic checked kernel (tiny).
  launch_transpose(W1, BT1, ZDD, ZDD, 256, stream);
  launch_gemm_fast(zbuf, BT1, b1, z1, BB, ZDD, 256, 1, stream);
  launch_transpose(W2, BT2, HDD, HDD, 128, stream);
  launch_gemm_fast(z1, BT2, b2, z2, BB, HDD, 128, 1, stream);
  {
    const int tiles  = ((BB + 15) / 16) * ((OUTD + 15) / 16);
    const int blocks = (tiles + 7) / 8;
    gemm_wmma_gen<<<blocks, 256, 0, stream>>>(z2, W3, b3, out, BB, 128, OUTD, 0);
  }
}